// Model_26079041421856
// MI455X (gfx1250) — compile-verified
//
#include <hip/hip_runtime.h>

typedef __attribute__((ext_vector_type(2))) float v2f;
typedef __attribute__((ext_vector_type(8))) float v8f;

#define SEQ    8192
#define DMODEL 16
#define NHEAD  4
#define DHEAD  4
#define FFDIM  64

// ---------------------------------------------------------------------------
// V_WMMA_F32_16X16X4_F32 : D(16x16,f32) = A(16x4,f32) x B(4x16,f32) + C
// ---------------------------------------------------------------------------
__device__ __forceinline__ v8f wmma_16x16x4(v2f a, v2f b, v8f c) {
    return __builtin_amdgcn_wmma_f32_16x16x4_f32(
        false, a, false, b, (short)0, c, false, false);
}

// Hardware tanh (gfx1250 v_tanh_f32) if the builtin exists on this toolchain.
#if __has_builtin(__builtin_amdgcn_tanhf)
#define HAVE_HW_TANH 1
__device__ __forceinline__ float hw_tanh(float x) { return __builtin_amdgcn_tanhf(x); }
#elif __has_builtin(__builtin_amdgcn_tanh_f32)
#define HAVE_HW_TANH 1
__device__ __forceinline__ float hw_tanh(float x) { return __builtin_amdgcn_tanh_f32(x); }
#else
#define HAVE_HW_TANH 0
#endif

// ---------------------------------------------------------------------------
// Embedding + positional: x[s,d] = emb[w[s],d] + s*pos_w[d,0]
//                                  + cos(pi*s)*pos_w[d,1] + pos_b[d]
// ---------------------------------------------------------------------------
__global__ void embed_kernel(const int* __restrict__ w,
                             const float* __restrict__ emb,
                             const float* __restrict__ pos_w,
                             const float* __restrict__ pos_b,
                             float* __restrict__ X) {
    int i = blockIdx.x * blockDim.x + threadIdx.x;   // SEQ*DMODEL threads
    int s = i >> 4;
    int d = i & 15;
    float par = (s & 1) ? -1.0f : 1.0f;
    X[i] = emb[w[s] * DMODEL + d]
         + pos_w[d * 2 + 0] * (float)s
         + pos_w[d * 2 + 1] * par
         + pos_b[d];
}

// ---------------------------------------------------------------------------
// Generic tiny GEMM via chained WMMA 16x16x4:
//   Y[S,DOUT] = act( X[S,DIN] @ W^T + bias ) [+ resid],  W is [DOUT,DIN]
// One wave per 16-row tile; K split into DIN/4 chunks.
// ---------------------------------------------------------------------------
template<int DIN, int DOUT, bool RELU>
__global__ void gemm16_kernel(const float* __restrict__ X,
                              const float* __restrict__ W,
                              const float* __restrict__ bias,
                              const float* __restrict__ resid,
                              float* __restrict__ Y) {
    const int lane = threadIdx.x & 31;
    const int wave = (blockIdx.x * blockDim.x + threadIdx.x) >> 5;
    const int row0 = wave * 16;
    const int hi   = lane >> 4;
    const int lm   = lane & 15;

    constexpr int KC = DIN / 4;

    // A-layout: lane = M (row), VGPR j holds K = 4*c + 2*hi + j
    v2f a[KC];
#pragma unroll
    for (int c = 0; c < KC; ++c) {
        const float* p = X + (size_t)(row0 + lm) * DIN + 4 * c + 2 * hi;
        a[c] = *(const v2f*)p;
    }

#pragma unroll
    for (int nt = 0; nt < DOUT / 16; ++nt) {
        const int n = nt * 16 + lm;
        const float bv = bias[n];
        v8f c = {bv, bv, bv, bv, bv, bv, bv, bv};

#pragma unroll
        for (int cc = 0; cc < KC; ++cc) {
            // B-layout: lane = N, VGPR j holds K = 4*cc + 2*hi + j
            const float* p = W + (size_t)n * DIN + 4 * cc + 2 * hi;
            v2f b = *(const v2f*)p;
            c = wmma_16x16x4(a[cc], b, c);
        }

        // C layout: VGPR r -> row M = r + 8*hi, col N = lm
#pragma unroll
        for (int r = 0; r < 8; ++r) {
            float v = c[r];
            if (RELU) v = v > 0.0f ? v : 0.0f;
            const int srow = row0 + r + 8 * hi;
            if (resid) v += resid[(size_t)srow * DOUT + n];
            Y[(size_t)srow * DOUT + n] = v;
        }
    }
}

// ---------------------------------------------------------------------------
// Streaming sigmoid attention; one wave = (head, 16-query tile).
// sigmoid(s/2) = 0.5*tanh(s/4) + 0.5; the 0.25 scale is folded into Q
// (linear), and sum_t v[t] is row-independent, so AV splits into a
// tanh-weighted per-row part + one shared v-sum accumulator.
// ---------------------------------------------------------------------------
__global__ void attn_kernel(const float* __restrict__ qkv,
                            float* __restrict__ O) {
    const int lane = threadIdx.x & 31;
    const int wave = (blockIdx.x * blockDim.x + threadIdx.x) >> 5;
    const int h  = wave & (NHEAD - 1);
    const int qt = wave >> 2;
    const int hi = lane >> 4;
    const int lm = lane & 15;

    // Q tile (A layout), pre-scaled so scores come out of WMMA ready to use
    const float* qp = qkv + (size_t)(qt * 16 + lm) * 48 + h * DHEAD + 2 * hi;
    v2f a = *(const v2f*)qp;
#if HAVE_HW_TANH
    a *= 0.25f;     // 1/sqrt(DH) * 0.5 (tanh half-angle)
#else
    a *= 0.5f;      // 1/sqrt(DH)
#endif

    // Pointer induction over key tiles (stride 16 rows * 48 floats)
    const float* kp = qkv + (size_t)lm * 48 + DMODEL + h * DHEAD + 2 * hi;
    const float* vp = qkv + (size_t)lm * 48 + 2 * DMODEL + h * DHEAD;

    float po[8][4];
#pragma unroll
    for (int r = 0; r < 8; ++r)
#pragma unroll
        for (int d = 0; d < 4; ++d) po[r][d] = 0.0f;
#if HAVE_HW_TANH
    float vsum[4] = {0.0f, 0.0f, 0.0f, 0.0f};
#endif

#pragma unroll 4
    for (int t0 = 0; t0 < SEQ; t0 += 16) {
        const v2f    b  = *(const v2f*)kp;
        const float4 vr = *(const float4*)vp;
        kp += 16 * 48;
        vp += 16 * 48;

        v8f z = {0.f, 0.f, 0.f, 0.f, 0.f, 0.f, 0.f, 0.f};
        v8f sc = wmma_16x16x4(a, b, z);

#if HAVE_HW_TANH
#pragma unroll
        for (int r = 0; r < 8; ++r) {
            float t = hw_tanh(sc[r]);
            po[r][0] += t * vr.x;
            po[r][1] += t * vr.y;
            po[r][2] += t * vr.z;
            po[r][3] += t * vr.w;
        }
        vsum[0] += vr.x; vsum[1] += vr.y; vsum[2] += vr.z; vsum[3] += vr.w;
#else
#pragma unroll
        for (int r = 0; r < 8; ++r) {
            float attn = __builtin_amdgcn_rcpf(1.0f + __expf(-sc[r]));
            po[r][0] += attn * vr.x;
            po[r][1] += attn * vr.y;
            po[r][2] += attn * vr.z;
            po[r][3] += attn * vr.w;
        }
#endif
    }

    // Reduce over the 16 lanes of each half
#pragma unroll
    for (int m = 1; m < 16; m <<= 1) {
#pragma unroll
        for (int r = 0; r < 8; ++r)
#pragma unroll
            for (int d = 0; d < 4; ++d)
                po[r][d] += __shfl_xor(po[r][d], m, 32);
#if HAVE_HW_TANH
#pragma unroll
        for (int d = 0; d < 4; ++d)
            vsum[d] += __shfl_xor(vsum[d], m, 32);
#endif
    }

    if (lm == 0) {
#pragma unroll
        for (int r = 0; r < 8; ++r) {
            const int srow = qt * 16 + r + 8 * hi;
#if HAVE_HW_TANH
            float4 o4 = make_float4(0.5f * (po[r][0] + vsum[0]),
                                    0.5f * (po[r][1] + vsum[1]),
                                    0.5f * (po[r][2] + vsum[2]),
                                    0.5f * (po[r][3] + vsum[3]));
#else
            float4 o4 = make_float4(po[r][0], po[r][1], po[r][2], po[r][3]);
#endif
            *(float4*)(O + (size_t)srow * DMODEL + h * DHEAD) = o4;
        }
    }
}

// ---------------------------------------------------------------------------
// Row LayerNorm over D=16 (vectorized b128 row access), one thread per row.
// ---------------------------------------------------------------------------
__global__ void ln_kernel(const float* __restrict__ in,
                          const float* __restrict__ g,
                          const float* __restrict__ b,
                          float* __restrict__ out) {
    const int s = blockIdx.x * blockDim.x + threadIdx.x;
    const float4* rp = (const float4*)(in + (size_t)s * 16);
    float4 q[4];
#pragma unroll
    for (int i = 0; i < 4; ++i) q[i] = rp[i];

    float v[16];
#pragma unroll
    for (int i = 0; i < 4; ++i) {
        v[4 * i + 0] = q[i].x; v[4 * i + 1] = q[i].y;
        v[4 * i + 2] = q[i].z; v[4 * i + 3] = q[i].w;
    }
    float mu = 0.0f;
#pragma unroll
    for (int d = 0; d < 16; ++d) mu += v[d];
    mu *= (1.0f / 16.0f);
    float var = 0.0f;
#pragma unroll
    for (int d = 0; d < 16; ++d) { float t = v[d] - mu; var += t * t; }
    var *= (1.0f / 16.0f);
    const float inv = rsqrtf(var + 1e-5f);

    float4* op = (float4*)(out + (size_t)s * 16);
#pragma unroll
    for (int i = 0; i < 4; ++i) {
        float4 o;
        o.x = (v[4 * i + 0] - mu) * inv * g[4 * i + 0] + b[4 * i + 0];
        o.y = (v[4 * i + 1] - mu) * inv * g[4 * i + 1] + b[4 * i + 1];
        o.z = (v[4 * i + 2] - mu) * inv * g[4 * i + 2] + b[4 * i + 2];
        o.w = (v[4 * i + 3] - mu) * inv * g[4 * i + 3] + b[4 * i + 3];
        op[i] = o;
    }
}

// ---------------------------------------------------------------------------
// Head: out[0] = x[SEQ-1] . head_w + head_b
// ---------------------------------------------------------------------------
__global__ void head_kernel(const float* __restrict__ X,
                            const float* __restrict__ hw,
                            const float* __restrict__ hb,
                            float* __restrict__ out) {
    const int l = threadIdx.x;
    float p = (l < 16) ? X[(size_t)(SEQ - 1) * DMODEL + l] * hw[l] : 0.0f;
#pragma unroll
    for (int m = 1; m < 32; m <<= 1) p += __shfl_xor(p, m, 32);
    if (l == 0) out[0] = p + hb[0];
}

// ---------------------------------------------------------------------------
extern "C" void kernel_launch(void* const* d_in, const int* in_sizes, int n_in,
                              void* d_out, int out_size, void* d_ws, size_t ws_size,
                              hipStream_t stream) {
    (void)in_sizes; (void)n_in; (void)out_size; (void)ws_size;

    const int*   w     = (const int*)  d_in[0];
    const float* emb   = (const float*)d_in[1];
    const float* pos_w = (const float*)d_in[2];
    const float* pos_b = (const float*)d_in[3];
    const float* hw    = (const float*)d_in[28];
    const float* hb    = (const float*)d_in[29];

    float* ws  = (float*)d_ws;
    float* X   = ws;                       // [S,16]  activations
    float* QKV = X   + (size_t)SEQ * 16;   // [S,48]
    float* AO  = QKV + (size_t)SEQ * 48;   // [S,16]  attention output
    float* T   = AO  + (size_t)SEQ * 16;   // [S,16]  temp (pre-LN)
    float* H   = T   + (size_t)SEQ * 16;   // [S,64]  FFN hidden
    float* out = (float*)d_out;

    embed_kernel<<<(SEQ * DMODEL) / 256, 256, 0, stream>>>(w, emb, pos_w, pos_b, X);

    for (int l = 0; l < 2; ++l) {
        const int base = 4 + 12 * l;
        const float* in_w  = (const float*)d_in[base + 0];
        const float* in_b  = (const float*)d_in[base + 1];
        const float* out_w = (const float*)d_in[base + 2];
        const float* out_b = (const float*)d_in[base + 3];
        const float* ff1_w = (const float*)d_in[base + 4];
        const float* ff1_b = (const float*)d_in[base + 5];
        const float* ff2_w = (const float*)d_in[base + 6];
        const float* ff2_b = (const float*)d_in[base + 7];
        const float* n1_g  = (const float*)d_in[base + 8];
        const float* n1_b  = (const float*)d_in[base + 9];
        const float* n2_g  = (const float*)d_in[base + 10];
        const float* n2_b  = (const float*)d_in[base + 11];

        gemm16_kernel<16, 48, false><<<64, 256, 0, stream>>>(X, in_w, in_b, nullptr, QKV);
        attn_kernel<<<256, 256, 0, stream>>>(QKV, AO);
        gemm16_kernel<16, 16, false><<<64, 256, 0, stream>>>(AO, out_w, out_b, X, T);
        ln_kernel<<<SEQ / 256, 256, 0, stream>>>(T, n1_g, n1_b, X);
        gemm16_kernel<16, 64, true><<<64, 256, 0, stream>>>(X, ff1_w, ff1_b, nullptr, H);
        gemm16_kernel<64, 16, false><<<64, 256, 0, stream>>>(H, ff2_w, ff2_b, X, T);
        ln_kernel<<<SEQ / 256, 256, 0, stream>>>(T, n2_g, n2_b, X);
    }

    head_kernel<<<1, 32, 0, stream>>>(X, hw, hb, out);
}